// Flow_Attention_80753975099921
// MI455X (gfx1250) — compile-verified
//
#include <hip/hip_runtime.h>
#include <hip/hip_bf16.h>
#include <math.h>

// Problem dims (fixed by reference)
#define BB 64
#define TK 512
#define NK 256
#define HH 512

typedef __attribute__((ext_vector_type(2))) float v2f;
typedef __attribute__((ext_vector_type(8))) float v8f;

// ---------------------------------------------------------------------------
// K0: zero a float buffer (scores accumulated with global atomics)
// ---------------------------------------------------------------------------
__global__ __launch_bounds__(256)
void zero_kernel(float* __restrict__ p, int n)
{
    const int i = blockIdx.x * 256 + threadIdx.x;
    if (i < n) p[i] = 0.0f;
}

// ---------------------------------------------------------------------------
// K1: dec_fea[b,h] = b_dec[h] + sum_k s_t_hat[b,k] * W_dec[k,h]   ([B,2H]@[2H,H])
// ---------------------------------------------------------------------------
__global__ __launch_bounds__(256)
void dec_fea_kernel(const float* __restrict__ s_t_hat,
                    const float* __restrict__ W_dec,
                    const float* __restrict__ b_dec,
                    float* __restrict__ dec)
{
    const int b = blockIdx.y;
    const int h = blockIdx.x * 256 + threadIdx.x;
    const float* srow = s_t_hat + (size_t)b * (2 * HH);
    float acc = b_dec[h];
    for (int k = 0; k < 2 * HH; ++k)
        acc += srow[k] * W_dec[(size_t)k * HH + h];
    dec[(size_t)b * HH + h] = acc;
}

// ---------------------------------------------------------------------------
// K2: fused projection GEMM + tanh + dot-with-v epilogue (WMMA f32 16x16x4).
//   score[b,r] += sum_{h in block-N} v[h]*tanh((X[b,r,:]@W)[h] + dec[b,h] + flow[b,r]*Wc[h])
//   128x128 C-tile per block: 16 waves in 4x4 grid, each wave owns 2x2 WMMA tiles.
//   K staged through LDS in 16-deep slabs (A: 128x16, B: 16x128) to amortize
//   weight reads ~8x vs a 16-row tile.  Scores combined via LDS ds_add_f32 then
//   one global atomicAdd per row (scores buffer pre-zeroed).
// ---------------------------------------------------------------------------
#define KAP 18    // A LDS row stride (pad: 18i mod 64 distinct for i<16, 8B aligned)
#define KBP 144   // B LDS row stride (2*144 mod 64 = 32 -> half-groups hit disjoint banks)

__global__ __launch_bounds__(512)
void fused_score_kernel(const float* __restrict__ X,     // [B,R,H]
                        const float* __restrict__ W,     // [H,H]
                        const float* __restrict__ dec,   // [B,H]
                        const float* __restrict__ flow,  // [B,R] or nullptr
                        const float* __restrict__ Wc,    // [H]   or nullptr
                        const float* __restrict__ vvec,  // [H]
                        float* __restrict__ scores,      // [B,R], pre-zeroed, atomically accumulated
                        int R)
{
    __shared__ float Als[128 * KAP];
    __shared__ float Bls[16 * KBP];
    __shared__ float s_score[128];

    const int tid  = threadIdx.x;
    const int row0 = blockIdx.x * 128;          // flattened (b*R + r); 128 | R so no batch cross
    const int n0   = blockIdx.y * 128;          // output-column base
    const int b    = row0 / R;

    if (tid < 128) s_score[tid] = 0.0f;

    const int wave = tid >> 5;
    const int wm   = wave >> 2;                 // 0..3 : which 32-row stripe
    const int wn   = wave & 3;                  // 0..3 : which 32-col stripe
    const int lane = tid & 31;
    const int hl   = lane >> 4;                 // half: k-pair / row-group selector
    const int r16  = lane & 15;

    // staging coordinates (one float4 of A and one of B per thread per slab)
    const int ar = tid >> 2, ac = (tid & 3) * 4;     // A: 128 rows x 4 chunks
    const int bk = tid >> 5, bc = (tid & 31) * 4;    // B: 16 rows x 32 chunks

    v8f acc[2][2] = {};

    for (int kb = 0; kb < HH; kb += 16) {
        // ---- stage A slab [128 x 16] and B slab [16 x 128] into LDS ----
        {
            const float4 av = *(const float4*)(X + (size_t)(row0 + ar) * HH + kb + ac);
            float* ad = &Als[ar * KAP + ac];
            ad[0] = av.x; ad[1] = av.y; ad[2] = av.z; ad[3] = av.w;

            const float4 bv = *(const float4*)(W + (size_t)(kb + bk) * HH + n0 + bc);
            float* bd = &Bls[bk * KBP + bc];
            bd[0] = bv.x; bd[1] = bv.y; bd[2] = bv.z; bd[3] = bv.w;

            if (kb + 16 < HH) {  // overlap next slab fetch with this slab's WMMAs
                __builtin_prefetch(X + (size_t)(row0 + ar) * HH + kb + 16 + ac, 0, 3);
                __builtin_prefetch(W + (size_t)(kb + 16 + bk) * HH + n0 + bc, 0, 3);
            }
        }
        __syncthreads();

        // ---- 4 k-steps x (2x2) WMMA tiles ----
#pragma unroll
        for (int ks = 0; ks < 4; ++ks) {
            const int kl = ks * 4 + 2 * hl;     // lane's k-pair within the slab
            v2f af[2], bf[2];
#pragma unroll
            for (int mi = 0; mi < 2; ++mi) {
                const float* ap = &Als[(wm * 32 + mi * 16 + r16) * KAP + kl];
                af[mi].x = ap[0]; af[mi].y = ap[1];
            }
#pragma unroll
            for (int ni = 0; ni < 2; ++ni) {
                const float* bp = &Bls[kl * KBP + wn * 32 + ni * 16 + r16];
                bf[ni].x = bp[0]; bf[ni].y = bp[KBP];
            }
#pragma unroll
            for (int mi = 0; mi < 2; ++mi)
#pragma unroll
                for (int ni = 0; ni < 2; ++ni)
                    acc[mi][ni] = __builtin_amdgcn_wmma_f32_16x16x4_f32(
                        false, af[mi], false, bf[ni], (short)0, acc[mi][ni], false, false);
        }
        __syncthreads();
    }

    // ---- epilogue: + dec_fea + flow*Wc, tanh, * v, per-row partial sums ----
    float rowsum[2][8];
#pragma unroll
    for (int mi = 0; mi < 2; ++mi)
#pragma unroll
        for (int i = 0; i < 8; ++i) rowsum[mi][i] = 0.0f;

#pragma unroll
    for (int ni = 0; ni < 2; ++ni) {
        const int ncol  = n0 + wn * 32 + ni * 16 + r16;
        const float dv  = dec[(size_t)b * HH + ncol];
        const float wcv = Wc ? Wc[ncol] : 0.0f;
        const float vv  = vvec[ncol];
#pragma unroll
        for (int mi = 0; mi < 2; ++mi) {
#pragma unroll
            for (int i = 0; i < 8; ++i) {
                const int rowg = row0 + wm * 32 + mi * 16 + 8 * hl + i;  // flattened b*R+r
                const float fl = flow ? flow[rowg] : 0.0f;
                rowsum[mi][i] += vv * tanhf(acc[mi][ni][i] + dv + fl * wcv);
            }
        }
    }

    // reduce over the 16 columns held by this half-group (xor <16 keeps bit4)
#pragma unroll
    for (int m = 1; m < 16; m <<= 1)
#pragma unroll
        for (int mi = 0; mi < 2; ++mi)
#pragma unroll
            for (int i = 0; i < 8; ++i)
                rowsum[mi][i] += __shfl_xor(rowsum[mi][i], m, 32);

    if (r16 == 0) {
#pragma unroll
        for (int mi = 0; mi < 2; ++mi)
#pragma unroll
            for (int i = 0; i < 8; ++i)
                atomicAdd(&s_score[wm * 32 + mi * 16 + 8 * hl + i], rowsum[mi][i]);
    }
    __syncthreads();
    if (tid < 128)
        atomicAdd(&scores[row0 + tid], s_score[tid]);   // combine across N-split blocks
}

// ---------------------------------------------------------------------------
// K3: masked renorm softmax.  out_i = e_i*m_i / sum_j e_j*m_j (softmax denom cancels)
// ---------------------------------------------------------------------------
__global__ __launch_bounds__(256)
void softmax_renorm_kernel(const float* __restrict__ scores,
                           const float* __restrict__ mask,
                           float* __restrict__ out, int R)
{
    __shared__ float red[256];
    const int b = blockIdx.x, tid = threadIdx.x;
    const float* s  = scores + (size_t)b * R;
    const float* mk = mask   + (size_t)b * R;
    float*       o  = out    + (size_t)b * R;

    float mx = -INFINITY;
    for (int i = tid; i < R; i += 256) mx = fmaxf(mx, s[i]);
    red[tid] = mx; __syncthreads();
    for (int st = 128; st > 0; st >>= 1) {
        if (tid < st) red[tid] = fmaxf(red[tid], red[tid + st]);
        __syncthreads();
    }
    mx = red[0]; __syncthreads();

    float acc = 0.0f;
    for (int i = tid; i < R; i += 256) acc += expf(s[i] - mx) * mk[i];
    red[tid] = acc; __syncthreads();
    for (int st = 128; st > 0; st >>= 1) {
        if (tid < st) red[tid] += red[tid + st];
        __syncthreads();
    }
    const float inv = 1.0f / red[0];
    for (int i = tid; i < R; i += 256) o[i] = expf(s[i] - mx) * mk[i] * inv;
}

// ---------------------------------------------------------------------------
// K4: gathered[b,t] = attn_node[b, node_to_token[b,t]]
// ---------------------------------------------------------------------------
__global__ __launch_bounds__(256)
void gather_kernel(const float* __restrict__ attn_node,
                   const int* __restrict__ n2t,
                   float* __restrict__ out)
{
    const int b = blockIdx.x;
    const int t = blockIdx.y * 256 + threadIdx.x;
    out[(size_t)b * TK + t] = attn_node[(size_t)b * NK + n2t[(size_t)b * TK + t]];
}

// ---------------------------------------------------------------------------
// K5: final_attn = 0.5*attn_tok + 0.5*attn_n2t
// ---------------------------------------------------------------------------
__global__ __launch_bounds__(256)
void combine_kernel(const float* __restrict__ a, const float* __restrict__ c,
                    float* __restrict__ out)
{
    const int i = blockIdx.x * 256 + threadIdx.x;
    out[i] = 0.5f * a[i] + 0.5f * c[i];
}

// ---------------------------------------------------------------------------
// K6: c_t[b,h] = sum_t final_attn[b,t] * enc_tok[b,t,h]
// ---------------------------------------------------------------------------
__global__ __launch_bounds__(256)
void ctx_kernel(const float* __restrict__ final_attn,
                const float* __restrict__ enc_tok,
                float* __restrict__ c_t)
{
    __shared__ float a[TK];
    const int b = blockIdx.x;
    const int h = blockIdx.y * 256 + threadIdx.x;
    for (int i = threadIdx.x; i < TK; i += 256)
        a[i] = final_attn[(size_t)b * TK + i];
    __syncthreads();
    const float* X = enc_tok + (size_t)b * TK * HH + h;
    float acc = 0.0f;
    for (int t = 0; t < TK; ++t) acc += a[t] * X[(size_t)t * HH];
    c_t[(size_t)b * HH + h] = acc;
}

// ---------------------------------------------------------------------------
// K7: flow_out = (z + z@G + (z@G)@G) * 0.33333
// ---------------------------------------------------------------------------
__global__ __launch_bounds__(256)
void flow_kernel(const float* __restrict__ attn_node,
                 const float* __restrict__ graph,
                 float* __restrict__ out)
{
    __shared__ float z[NK];
    __shared__ float one[NK];
    const int b = blockIdx.x, m = threadIdx.x;
    z[m] = attn_node[(size_t)b * NK + m];
    __syncthreads();
    const float* G = graph + (size_t)b * NK * NK;
    float acc = 0.0f;
    for (int n = 0; n < NK; ++n) acc += z[n] * G[(size_t)n * NK + m];   // coalesced over m
    one[m] = acc; __syncthreads();
    float acc2 = 0.0f;
    for (int n = 0; n < NK; ++n) acc2 += one[n] * G[(size_t)n * NK + m];
    out[(size_t)b * NK + m] = (z[m] + acc + acc2) * 0.33333f;
}

// ---------------------------------------------------------------------------
extern "C" void kernel_launch(void* const* d_in, const int* in_sizes, int n_in,
                              void* d_out, int out_size, void* d_ws, size_t ws_size,
                              hipStream_t stream)
{
    const float* s_t_hat   = (const float*)d_in[0];   // [B,2H]
    const float* enc_tok   = (const float*)d_in[1];   // [B,TK,H]
    const float* enc_node  = (const float*)d_in[2];   // [B,NK,H]
    const int*   n2t       = (const int*)  d_in[3];   // [B,TK]
    const float* mask_tok  = (const float*)d_in[4];   // [B,TK]
    const float* mask_node = (const float*)d_in[5];   // [B,NK]
    const float* graph     = (const float*)d_in[6];   // [B,NK,NK]
    const float* flow      = (const float*)d_in[7];   // [B,NK]
    const float* W_c       = (const float*)d_in[8];   // [H]
    const float* W_tok     = (const float*)d_in[9];   // [H,H]
    const float* W_node    = (const float*)d_in[10];  // [H,H]
    const float* W_dec     = (const float*)d_in[11];  // [2H,H]
    const float* b_dec     = (const float*)d_in[12];  // [H]
    const float* v_tok     = (const float*)d_in[13];  // [H]
    const float* v_node    = (const float*)d_in[14];  // [H]

    float* out = (float*)d_out;
    float* out_ct    = out;                               // [B,H]   32768
    float* out_final = out + BB * HH;                     // [B,TK]  32768
    float* out_anode = out + BB * HH + BB * TK;           // [B,NK]  16384
    float* out_flow  = out + BB * HH + BB * TK + BB * NK; // [B,NK]  16384

    // workspace layout (floats) -- scores_node/scores_tok contiguous for one zero pass
    float* ws = (float*)d_ws;
    float* dec_fea     = ws;                     // B*H  = 32768
    float* scores_node = dec_fea + BB * HH;      // B*NK = 16384
    float* scores_tok  = scores_node + BB * NK;  // B*TK = 32768
    float* attn_tok    = scores_tok + BB * TK;   // B*TK
    float* gathered    = attn_tok + BB * TK;     // B*TK
    float* attn_n2t    = gathered + BB * TK;     // B*TK

    // K0: zero the atomically-accumulated score buffers (49152 floats, contiguous)
    {
        const int n = BB * NK + BB * TK;
        zero_kernel<<<(n + 255) / 256, 256, 0, stream>>>(scores_node, n);
    }

    // K1: decoder projection
    dec_fea_kernel<<<dim3(HH / 256, BB), 256, 0, stream>>>(s_t_hat, W_dec, b_dec, dec_fea);

    // K2: fused WMMA GEMM + tanh-dot epilogues, 128x128 tiles, N split over blockIdx.y
    fused_score_kernel<<<dim3(BB * NK / 128, HH / 128), 512, 0, stream>>>(
        enc_node, W_node, dec_fea, flow, W_c, v_node, scores_node, NK);
    fused_score_kernel<<<dim3(BB * TK / 128, HH / 128), 512, 0, stream>>>(
        enc_tok, W_tok, dec_fea, nullptr, nullptr, v_tok, scores_tok, TK);

    // K3: masked-renorm softmaxes
    softmax_renorm_kernel<<<BB, 256, 0, stream>>>(scores_node, mask_node, out_anode, NK);
    softmax_renorm_kernel<<<BB, 256, 0, stream>>>(scores_tok, mask_tok, attn_tok, TK);

    // K4: gather node attention onto tokens, then renorm-softmax it
    gather_kernel<<<dim3(BB, TK / 256), 256, 0, stream>>>(out_anode, n2t, gathered);
    softmax_renorm_kernel<<<BB, 256, 0, stream>>>(gathered, mask_tok, attn_n2t, TK);

    // K5: mix
    combine_kernel<<<BB * TK / 256, 256, 0, stream>>>(attn_tok, attn_n2t, out_final);

    // K6: context vector
    ctx_kernel<<<dim3(BB, HH / 256), 256, 0, stream>>>(out_final, enc_tok, out_ct);

    // K7: graph diffusion
    flow_kernel<<<BB, NK, 0, stream>>>(out_anode, graph, out_flow);
}